// RnntLoss_58153857188540
// MI455X (gfx1250) — compile-verified
//
#include <hip/hip_runtime.h>

#define Bb 16
#define Tt 200
#define Uu 50
#define Vv 1024
#define NEGINF (-1e9f)

typedef __attribute__((ext_vector_type(2))) float v2f;
typedef __attribute__((ext_vector_type(8))) float v8f;

__device__ __forceinline__ float logaddexp_f(float a, float b) {
    float m = fmaxf(a, b);
    float n = fminf(a, b);
    // exp underflows to 0 for very negative (n-m), giving m exactly.
    return m + log1pf(__expf(n - m));
}

// ---------------------------------------------------------------------------
// Phase 1: fused log-softmax -> blank/label log-probs.
// One wave32 per (b,t,u) row of 1024 logits. Memory-bound: 668 MB streamed.
// Cross-lane sum of exp-partials uses V_WMMA_F32_16X16X4_F32 (A=ones) as the
// first reduction stage (32 lanes -> 16), then 4 xor-shuffles.
// ---------------------------------------------------------------------------
__global__ void __launch_bounds__(256)
rnnt_lse_kernel(const float* __restrict__ logits,
                const int* __restrict__ targets,
                float* __restrict__ blank_lp,
                float* __restrict__ label_lp) {
    const int wave = blockIdx.x * (blockDim.x >> 5) + (threadIdx.x >> 5);
    const int lane = threadIdx.x & 31;
    const int row  = wave;                       // 0 .. B*T*(U+1)-1 exactly
    const int u    = row % (Uu + 1);
    const int bt   = row / (Uu + 1);
    const int b    = bt / Tt;
    const int t    = bt % Tt;

    const float* base = logits + (size_t)row * Vv;
    const float4* b4  = (const float4*)base;

    float x[32];
#pragma unroll
    for (int i = 0; i < 8; ++i) {
        float4 v = b4[i * 32 + lane];            // coalesced 512B per 32 lanes
        x[i * 4 + 0] = v.x; x[i * 4 + 1] = v.y;
        x[i * 4 + 2] = v.z; x[i * 4 + 3] = v.w;
    }

    // row max
    float m = x[0];
#pragma unroll
    for (int i = 1; i < 32; ++i) m = fmaxf(m, x[i]);
#pragma unroll
    for (int off = 16; off > 0; off >>= 1)
        m = fmaxf(m, __shfl_xor(m, off, 32));

    // per-lane partial sum of exp(x - m)
    float p = 0.f;
#pragma unroll
    for (int i = 0; i < 32; ++i) p += __expf(x[i] - m);

    // WMMA reduction stage: D[i,j] = sum_k A[i,k]*B[k,j] with A == 1
    //   B layout (4x16, f32): VGPR0 lanes0-15 = K0 row, lanes16-31 = K1 row;
    //   VGPR1 lanes0-15 = K2 row, lanes16-31 = K3 row.
    //   With both B VGPRs = p:  D[i,j] = 2*(p[j] + p[j+16])  (indep. of i).
    //   D VGPR0 lane l holds that value for j = l%16.
    v2f av; av.x = 1.f; av.y = 1.f;
    v2f bv; bv.x = p;   bv.y = p;
    v8f cz = {};
    v8f d = __builtin_amdgcn_wmma_f32_16x16x4_f32(
        /*neg_a=*/false, av, /*neg_b=*/false, bv,
        /*c_mod=*/(short)0, cz, /*reuse_a=*/false, /*reuse_b=*/false);

    float s = d[0];
#pragma unroll
    for (int off = 1; off <= 8; off <<= 1)
        s += __shfl_xor(s, off, 32);
    s *= 0.5f;                                    // undo the factor-2 from WMMA

    const float lse = m + __logf(s);

    if (lane == 0) {
        blank_lp[row] = base[0] - lse;            // blank_label == 0
        if (u < Uu) {
            const int tgt = targets[b * Uu + u];  // in [1, V)
            label_lp[(size_t)(b * Tt + t) * Uu + u] = base[tgt] - lse;
        }
    }
}

// ---------------------------------------------------------------------------
// Phase 2: alpha trellis via anti-diagonal wavefront.
// One 64-thread block per batch element; frontier alpha[u] lives in LDS.
// At diagonal d, thread u updates cell (t = d-u, u):
//   alpha[t,u] = logaddexp(alpha[t-1,u] + blank[t-1,u],
//                          alpha[t,u-1] + label[t,u-1])
// A[u] holds alpha[t-1,u]; A[u-1] already holds alpha[t,u-1] (updated at d-1).
// ---------------------------------------------------------------------------
__global__ void __launch_bounds__(64)
rnnt_dp_kernel(const float* __restrict__ blank_lp,
               const float* __restrict__ label_lp,
               const int* __restrict__ logit_lengths,
               const int* __restrict__ target_lengths,
               float* __restrict__ ll) {
    const int b = blockIdx.x;
    const int u = threadIdx.x;
    __shared__ float A[Uu + 1];

    const float* blk = blank_lp + (size_t)b * Tt * (Uu + 1);
    const float* lab = label_lp + (size_t)b * Tt * Uu;
    const int t_idx = logit_lengths[b] - 1;
    const int u_idx = target_lengths[b];

    // t = 0 row: pure label prefix sums
    if (u == 0) {
        float a = 0.f;
        A[0] = 0.f;
        for (int j = 1; j <= Uu; ++j) { a += lab[j - 1]; A[j] = a; }
    }
    __syncthreads();

    if (t_idx == 0 && u == 0)                     // defensive; lengths >= T/2
        ll[b] = A[u_idx] + blk[u_idx];

    for (int dgn = 1; dgn <= (Tt - 1) + Uu; ++dgn) {
        const int t = dgn - u;
        const bool active = (u <= Uu) && (t >= 1) && (t <= Tt - 1);
        float aUp = 0.f, aLeft = NEGINF;
        if (active) {
            aUp = A[u];
            if (u > 0) aLeft = A[u - 1];
        }
        __syncthreads();
        if (active) {
            const float vert = aUp + blk[(t - 1) * (Uu + 1) + u];
            const float diag = (u > 0) ? (aLeft + lab[t * Uu + (u - 1)])
                                       : NEGINF;
            const float a = logaddexp_f(vert, diag);
            A[u] = a;
            if (t == t_idx && u == u_idx)
                ll[b] = a + blk[t * (Uu + 1) + u];
        }
        __syncthreads();
    }
}

// ---------------------------------------------------------------------------
// Phase 3: loss = -mean(ll)
// ---------------------------------------------------------------------------
__global__ void rnnt_finalize_kernel(const float* __restrict__ ll,
                                     float* __restrict__ out) {
    if (threadIdx.x == 0) {
        float s = 0.f;
        for (int b = 0; b < Bb; ++b) s += ll[b];
        out[0] = -s / (float)Bb;
    }
}

extern "C" void kernel_launch(void* const* d_in, const int* in_sizes, int n_in,
                              void* d_out, int out_size, void* d_ws, size_t ws_size,
                              hipStream_t stream) {
    const float* logits         = (const float*)d_in[0];
    const int*   targets        = (const int*)d_in[1];
    const int*   logit_lengths  = (const int*)d_in[2];
    const int*   target_lengths = (const int*)d_in[3];
    float* out = (float*)d_out;

    float* ws       = (float*)d_ws;
    float* blank_lp = ws;                                   // B*T*(U+1)
    float* label_lp = ws + Bb * Tt * (Uu + 1);              // B*T*U
    float* ll       = label_lp + Bb * Tt * Uu;              // B

    const int rows = Bb * Tt * (Uu + 1);                    // 163200
    // 256 threads = 8 wave32 per block, one row per wave; rows % 8 == 0
    rnnt_lse_kernel<<<rows / 8, 256, 0, stream>>>(logits, targets,
                                                  blank_lp, label_lp);
    rnnt_dp_kernel<<<Bb, 64, 0, stream>>>(blank_lp, label_lp,
                                          logit_lengths, target_lengths, ll);
    rnnt_finalize_kernel<<<1, 32, 0, stream>>>(ll, out);
}